// ImageTextContrastiveLossV2_75960791597441
// MI455X (gfx1250) — compile-verified
//
#include <hip/hip_runtime.h>
#include <hip/hip_bf16.h>

typedef __attribute__((ext_vector_type(2))) float v2f;
typedef __attribute__((ext_vector_type(8))) float v8f;
typedef __attribute__((ext_vector_type(4))) unsigned int u32x4;
typedef __attribute__((ext_vector_type(8))) int i32x8;
typedef __attribute__((ext_vector_type(4))) int i32x4;

#define DDIM 256
#define NEG_BIG -3.0e38f
#define CFENCE() asm volatile("" ::: "memory")

// ---------------------------------------------------------------------------
// TDM: async-copy a 16x256 fp32 tile (row stride 256 elements) to LDS.
// D# per CDNA5 ISA ch.8: group0 = count/lds_addr/global_addr/type,
// group1 = data_size + dims/strides, groups 2&3 zero (2-D tile).
// Tracked with TENSORcnt; completes in order with other TDM ops of this wave.
// This toolchain exposes the 6-arg builtin (extra zero descriptor group).
// ---------------------------------------------------------------------------
__device__ __forceinline__ void tdm_load_tile_16x256(const void* tile,
                                                     unsigned lds_off) {
    unsigned long long ga = (unsigned long long)tile;
    u32x4 g0;
    g0[0] = 1u;                                     // count=1 (valid), user mode
    g0[1] = lds_off;                                // lds_addr (bytes)
    g0[2] = (unsigned)(ga & 0xFFFFFFFFull);         // global_addr[31:0]
    g0[3] = (unsigned)((ga >> 32) & 0x01FFFFFFull)  // global_addr[56:32]
            | (2u << 30);                           // type=2 ("image")
    i32x8 g1;
    g1[0] = (int)(2u << 16);                        // data_size=2 -> 4 bytes
    g1[1] = (int)(256u << 16);                      // tensor_dim0 = 256
    g1[2] = (int)(16u << 16);                       // tensor_dim1 = 16
    g1[3] = (int)(256u << 16);                      // tile_dim0 = 256
    g1[4] = 16;                                     // tile_dim1 = 16 (dim2=0)
    g1[5] = 256;                                    // tensor_dim0_stride = 256
    g1[6] = 0;
    g1[7] = 0;
    i32x4 z4 = {0, 0, 0, 0};
    i32x8 z8 = {0, 0, 0, 0, 0, 0, 0, 0};
    __builtin_amdgcn_tensor_load_to_lds(g0, g1, z4, z4, z8, 0);
}

// ---------------------------------------------------------------------------
// One wave (32 lanes) per row: inv_norm[r] = 1 / max(||x_r||, 1e-12)
// ---------------------------------------------------------------------------
__global__ void row_inv_norm_kernel(const float* __restrict__ x,
                                    float* __restrict__ inv, int nrows) {
    int wave = (blockIdx.x * blockDim.x + threadIdx.x) >> 5;
    int lane = threadIdx.x & 31;
    if (wave >= nrows) return;
    const float* row = x + (size_t)wave * DDIM;
    float s = 0.f;
    #pragma unroll
    for (int i = lane; i < DDIM; i += 32) { float v = row[i]; s += v * v; }
    #pragma unroll
    for (int off = 16; off >= 1; off >>= 1) s += __shfl_xor(s, off, 32);
    if (lane == 0) inv[wave] = 1.0f / fmaxf(sqrtf(s), 1e-12f);
}

// ---------------------------------------------------------------------------
// One wave per row: y_r = x_r / max(||x_r||, 1e-12)   (prompt normalization)
// ---------------------------------------------------------------------------
__global__ void normalize_rows_kernel(const float* __restrict__ x,
                                      float* __restrict__ y, int nrows) {
    int wave = (blockIdx.x * blockDim.x + threadIdx.x) >> 5;
    int lane = threadIdx.x & 31;
    if (wave >= nrows) return;
    const float* row = x + (size_t)wave * DDIM;
    float* out = y + (size_t)wave * DDIM;
    float s = 0.f;
    #pragma unroll
    for (int i = lane; i < DDIM; i += 32) { float v = row[i]; s += v * v; }
    #pragma unroll
    for (int off = 16; off >= 1; off >>= 1) s += __shfl_xor(s, off, 32);
    float sc = 1.0f / fmaxf(sqrtf(s), 1e-12f);
    #pragma unroll
    for (int i = lane; i < DDIM; i += 32) out[i] = row[i] * sc;
}

// ---------------------------------------------------------------------------
// Main fused kernel: one wave per 16-image tile.
//   A tile (raw image rows) and B tiles (normalized prompts) staged via the
//   Tensor Data Mover (double-buffered B), 64x v_wmma_f32_16x16x4_f32 per
//   16x16 sim tile, image inv-norm folded into the epilogue scale.
//   Per-row online LSE of logits = (pos ? sim/T : 0), num_pos, sum_pos(sim/T).
// ---------------------------------------------------------------------------
__global__ __launch_bounds__(32)
void attr_loss_kernel(const float* __restrict__ img,
                      const float* __restrict__ inv_norm,
                      const float* __restrict__ zp,
                      const int* __restrict__ labels,
                      int C,
                      float* __restrict__ partial_sum,
                      float* __restrict__ partial_cnt) {
    __shared__ float Ash[16 * DDIM];
    __shared__ float Bsh[2][16 * DDIM];

    const int lane = threadIdx.x;           // 0..31
    const int row0 = blockIdx.x * 16;
    const int mrow = lane & 15;             // A: M index / B: N index in tile
    const int kgrp = (lane >> 4) << 1;      // K sub-offset {0,2} by half-wave
    const int hw   = lane >> 4;
    const int ntiles = C >> 4;

    const unsigned ashOff = (unsigned)(unsigned long long)(void*)&Ash[0];
    const unsigned bshOff[2] = {
        (unsigned)(unsigned long long)(void*)&Bsh[0][0],
        (unsigned)(unsigned long long)(void*)&Bsh[1][0]
    };

    // Kick off A tile + first B tile on the TDM.
    tdm_load_tile_16x256(img + (size_t)row0 * DDIM, ashOff);
    tdm_load_tile_16x256(zp, bshOff[0]);

    // Per-row scale = inv_norm / T (folded-out image normalization).
    float inva10[8];
    #pragma unroll
    for (int j = 0; j < 8; ++j)
        inva10[j] = inv_norm[row0 + j + 8 * hw] * 10.0f;

    // Per-lane per-row accumulators (row j+8*hw, column stream N = lane&15).
    float m[8], se[8], np[8], sp[8];
    #pragma unroll
    for (int j = 0; j < 8; ++j) { m[j] = NEG_BIG; se[j] = 0.f; np[j] = 0.f; sp[j] = 0.f; }

    for (int nt = 0; nt < ntiles; ++nt) {
        const int n0 = nt * 16;
        if (nt + 1 < ntiles) {
            CFENCE();
            // Prefetch next B tile into the other buffer, then wait until only
            // that prefetch is outstanding (=> A + current B are resident).
            tdm_load_tile_16x256(zp + (size_t)(nt + 1) * 16 * DDIM,
                                 bshOff[(nt + 1) & 1]);
            __builtin_amdgcn_s_wait_tensorcnt(1);
        } else {
            __builtin_amdgcn_s_wait_tensorcnt(0);
        }
        CFENCE();
        const float* Bcur = &Bsh[nt & 1][0];

        v8f acc = {};
        #pragma unroll
        for (int s = 0; s < DDIM / 4; ++s) {
            int kk = s * 4 + kgrp;
            v2f a = *(const v2f*)&Ash[mrow * DDIM + kk];
            v2f b = *(const v2f*)&Bcur[mrow * DDIM + kk];
            acc = __builtin_amdgcn_wmma_f32_16x16x4_f32(
                /*neg_a=*/false, a, /*neg_b=*/false, b,
                /*c_mod=*/(short)0, acc, /*reuse_a=*/false, /*reuse_b=*/false);
        }

        // Apply inv_norm/T, mask with labels, online log-sum-exp per row.
        #pragma unroll
        for (int j = 0; j < 8; ++j) {
            int r = row0 + j + 8 * hw;
            int c = n0 + (lane & 15);
            float s10 = acc[j] * inva10[j];             // sim / 0.1
            bool pos = labels[(size_t)r * C + c] > 0;
            float x = pos ? s10 : 0.0f;                 // faithful masked logit
            float mo = m[j];
            float mn = fmaxf(mo, x);
            se[j] = se[j] * __expf(mo - mn) + __expf(x - mn);
            m[j] = mn;
            if (pos) { np[j] += 1.0f; sp[j] += s10; }
        }
    }

    // Butterfly reduce across the 16-lane group holding each row's columns.
    #pragma unroll
    for (int j = 0; j < 8; ++j) {
        float mm = m[j], ss = se[j], nn = np[j], pp = sp[j];
        #pragma unroll
        for (int off = 8; off >= 1; off >>= 1) {
            float mo = __shfl_xor(mm, off, 32);
            float so = __shfl_xor(ss, off, 32);
            nn += __shfl_xor(nn, off, 32);
            pp += __shfl_xor(pp, off, 32);
            float mx = fmaxf(mm, mo);
            ss = ss * __expf(mm - mx) + so * __expf(mo - mx);
            mm = mx;
        }
        m[j] = mm; se[j] = ss; np[j] = nn; sp[j] = pp;
    }

    // Lanes 0 and 16 now hold final stats for rows j and j+8 respectively.
    float lsum = 0.f, lcnt = 0.f;
    if ((lane & 15) == 0) {
        #pragma unroll
        for (int j = 0; j < 8; ++j) {
            if (np[j] > 0.f) {
                float lse = m[j] + __logf(se[j]);
                lsum += (np[j] * lse - sp[j]) / (np[j] + 1e-8f);
                lcnt += 1.0f;
            }
        }
    }
    lsum += __shfl_xor(lsum, 16, 32);
    lcnt += __shfl_xor(lcnt, 16, 32);
    if (lane == 0) {
        partial_sum[blockIdx.x] = lsum;
        partial_cnt[blockIdx.x] = lcnt;
    }
}

// ---------------------------------------------------------------------------
// Deterministic final reduction + scalar combine (reference semantics).
// ---------------------------------------------------------------------------
__device__ float block_sum256(float v, float* sh) {
    int t = threadIdx.x;
    sh[t] = v;
    __syncthreads();
    #pragma unroll
    for (int s = 128; s >= 1; s >>= 1) {
        if (t < s) sh[t] += sh[t + s];
        __syncthreads();
    }
    float r = sh[0];
    __syncthreads();
    return r;
}

__global__ void finalize_kernel(const float* __restrict__ ps1,
                                const float* __restrict__ pc1,
                                const float* __restrict__ ps2,
                                const float* __restrict__ pc2,
                                int nblocks, float* __restrict__ out) {
    __shared__ float sh[256];
    int t = threadIdx.x;
    float a = 0.f, b = 0.f, c = 0.f, d = 0.f;
    for (int i = t; i < nblocks; i += 256) {
        a += ps1[i]; b += pc1[i]; c += ps2[i]; d += pc2[i];
    }
    a = block_sum256(a, sh);
    b = block_sum256(b, sh);
    c = block_sum256(c, sh);
    d = block_sum256(d, sh);
    if (t == 0) {
        float loss1 = a / fmaxf(b, 1.0f);
        float loss2 = c / fmaxf(d, 1.0f);
        float v1 = (b > 0.f) ? 1.0f : 0.0f;
        float v2 = (d > 0.f) ? 1.0f : 0.0f;
        float num_valid = v1 + v2;
        float total = loss1 * v1 + loss2 * v2;
        out[0] = (num_valid > 0.f) ? (total / fmaxf(num_valid, 1.0f)) : 0.0f;
    }
}

// ---------------------------------------------------------------------------
extern "C" void kernel_launch(void* const* d_in, const int* in_sizes, int n_in,
                              void* d_out, int out_size, void* d_ws, size_t ws_size,
                              hipStream_t stream) {
    const float* img  = (const float*)d_in[0];
    const float* prc  = (const float*)d_in[1];
    const float* prs  = (const float*)d_in[2];
    const int*   labc = (const int*)d_in[3];
    const int*   labs = (const int*)d_in[4];

    const int B  = in_sizes[0] / DDIM;   // 16384
    const int C1 = in_sizes[1] / DDIM;   // 1024
    const int C2 = in_sizes[2] / DDIM;   // 512
    const int nb = B / 16;

    float* w    = (float*)d_ws;
    float* inv  = w;                          // B
    float* zp1  = inv + B;                    // C1*DDIM
    float* zp2  = zp1 + (size_t)C1 * DDIM;    // C2*DDIM
    float* ps1  = zp2 + (size_t)C2 * DDIM;    // nb
    float* pc1w = ps1 + nb;                   // nb
    float* ps2  = pc1w + nb;                  // nb
    float* pc2w = ps2 + nb;                   // nb

    row_inv_norm_kernel<<<(B + 7) / 8, 256, 0, stream>>>(img, inv, B);
    normalize_rows_kernel<<<(C1 + 7) / 8, 256, 0, stream>>>(prc, zp1, C1);
    normalize_rows_kernel<<<(C2 + 7) / 8, 256, 0, stream>>>(prs, zp2, C2);

    attr_loss_kernel<<<nb, 32, 0, stream>>>(img, inv, zp1, labc, C1, ps1, pc1w);
    attr_loss_kernel<<<nb, 32, 0, stream>>>(img, inv, zp2, labs, C2, ps2, pc2w);

    finalize_kernel<<<1, 256, 0, stream>>>(ps1, pc1w, ps2, pc2w, nb, (float*)d_out);
}